// IterModel_66039417143523
// MI455X (gfx1250) — compile-verified
//
#include <hip/hip_runtime.h>
#include <hip/hip_bf16.h>
#include <math.h>

// ---------------------------------------------------------------------------
// Problem constants (match reference)
// ---------------------------------------------------------------------------
#define BB      256
#define LL      128
#define DD      768
#define TT      20480
#define NSLOTS  128
#define NLABELS 32
#define ITERS   3

typedef __attribute__((ext_vector_type(2))) float v2f;
typedef __attribute__((ext_vector_type(8))) float v8f;

#if defined(__AMDGCN__) && __has_builtin(__builtin_amdgcn_wmma_f32_16x16x4_f32)
#define HAVE_WMMA_F32 1
#else
#define HAVE_WMMA_F32 0
#endif

// ---------------------------------------------------------------------------
// Wave helpers (wave32)
// ---------------------------------------------------------------------------
__device__ __forceinline__ float wave_reduce_sum(float v) {
    for (int off = 16; off > 0; off >>= 1) v += __shfl_xor(v, off, 32);
    return v;
}

// ---------------------------------------------------------------------------
// Generic f32 WMMA GEMM: C[M,N] = A[M,K] @ B[K,N], all row-major.
// One wave computes a 16 x (16*NT) strip via V_WMMA_F32_16X16X4_F32, reusing
// each A fragment across NT column tiles (A traffic = roofline minimum).
// Grid must supply exactly (M/16) * (N/(16*NT)) waves (blockDim=256 -> 8/blk).
// ---------------------------------------------------------------------------
template <int NT>
__global__ void gemm_wmma_f32(const float* __restrict__ A,
                              const float* __restrict__ B,
                              float* __restrict__ C,
                              int M, int N, int K) {
    const int lane  = threadIdx.x & 31;
    const int wid   = blockIdx.x * (blockDim.x >> 5) + (threadIdx.x >> 5);
    const int tg    = N / (16 * NT);          // column groups
    const int m0    = (wid / tg) << 4;
    const int n0    = (wid % tg) * (16 * NT);
    const int lhalf = lane >> 4;              // 0: lanes 0-15, 1: lanes 16-31
    const int lmod  = lane & 15;

    v8f acc[NT];
#pragma unroll
    for (int j = 0; j < NT; ++j) acc[j] = (v8f){};

#if HAVE_WMMA_F32
    // A operand: lane holds row m0+lmod, K-pair (2*lhalf, 2*lhalf+1) per step
    const float* arow = A + (size_t)(m0 + lmod) * K + (lhalf << 1);
    // B operand: lane holds col n0+lmod+16j, K rows (2*lhalf, 2*lhalf+1)
    const float* brow = B + (size_t)(lhalf << 1) * N + (n0 + lmod);
    for (int k0 = 0; k0 < K; k0 += 4) {
        v2f a; a.x = arow[0]; a.y = arow[1];
        arow += 4;
#pragma unroll
        for (int j = 0; j < NT; ++j) {
            v2f b; b.x = brow[j * 16]; b.y = brow[(size_t)N + j * 16];
            acc[j] = __builtin_amdgcn_wmma_f32_16x16x4_f32(false, a, false, b,
                                                           (short)0, acc[j],
                                                           false, false);
        }
        brow += (size_t)4 * N;
    }
#else
    {
        const int row0 = m0 + (lhalf ? 8 : 0);
        for (int j = 0; j < NT; ++j) {
            for (int i = 0; i < 8; ++i) {
                const float* ar = A + (size_t)(row0 + i) * K;
                const float* bc = B + (n0 + j * 16 + lmod);
                float s = 0.f;
                for (int k = 0; k < K; ++k) s += ar[k] * bc[(size_t)k * N];
                acc[j][i] = s;
            }
        }
    }
#endif
    // C/D layout: VGPR i -> row m0 + i + 8*lhalf, col n0 + 16j + lmod
    const int row0 = m0 + (lhalf ? 8 : 0);
#pragma unroll
    for (int j = 0; j < NT; ++j) {
        float* crow = C + (size_t)row0 * N + (n0 + j * 16 + lmod);
#pragma unroll
        for (int i = 0; i < 8; ++i) crow[(size_t)i * N] = acc[j][i];
    }
}

// ---------------------------------------------------------------------------
// slot_output = [H | R_slot] @ W_slot, with R_slot[t,d] = F[sid[t],d]*Cp[t,d]
// synthesized on the fly (never materialized).  M=T, K=1536, N=128.
// One wave computes the full 16x128 output strip (8 accumulators), so the big
// streaming operands H/Cp are read exactly once (126 MB roofline minimum);
// W_slot (786 KB) stays L2-resident.
// ---------------------------------------------------------------------------
__global__ void gemm_slot_wmma(const float* __restrict__ H,
                               const float* __restrict__ Cp,
                               const float* __restrict__ F,
                               const int*   __restrict__ sid,
                               const float* __restrict__ W,   // [2D, 128]
                               float* __restrict__ C) {       // [T, 128]
    const int lane  = threadIdx.x & 31;
    const int wid   = blockIdx.x * 8 + (threadIdx.x >> 5);
    const int m0    = wid << 4;              // one column group (NT=8 covers N)
    const int lhalf = lane >> 4;
    const int lmod  = lane & 15;

    v8f acc[8];
#pragma unroll
    for (int j = 0; j < 8; ++j) acc[j] = (v8f){};

#if HAVE_WMMA_F32
    const int    t    = m0 + lmod;
    const int    b    = sid[t];
    const float* hrow = H  + (size_t)t * DD + (lhalf << 1);
    const float* crow = Cp + (size_t)t * DD + (lhalf << 1);
    const float* frow = F  + (size_t)b * DD + (lhalf << 1);
    const float* brow = W + (size_t)(lhalf << 1) * NSLOTS + lmod;
    // K in [0, 768): A from H
    for (int k0 = 0; k0 < DD; k0 += 4) {
        v2f a; a.x = hrow[0]; a.y = hrow[1];
        hrow += 4;
#pragma unroll
        for (int j = 0; j < 8; ++j) {
            v2f b2; b2.x = brow[j * 16]; b2.y = brow[NSLOTS + j * 16];
            acc[j] = __builtin_amdgcn_wmma_f32_16x16x4_f32(false, a, false, b2,
                                                           (short)0, acc[j],
                                                           false, false);
        }
        brow += 4 * NSLOTS;
    }
    // K in [768, 1536): A = F ⊙ Cp
    for (int k0 = 0; k0 < DD; k0 += 4) {
        v2f a; a.x = frow[0] * crow[0]; a.y = frow[1] * crow[1];
        frow += 4; crow += 4;
#pragma unroll
        for (int j = 0; j < 8; ++j) {
            v2f b2; b2.x = brow[j * 16]; b2.y = brow[NSLOTS + j * 16];
            acc[j] = __builtin_amdgcn_wmma_f32_16x16x4_f32(false, a, false, b2,
                                                           (short)0, acc[j],
                                                           false, false);
        }
        brow += 4 * NSLOTS;
    }
#else
    {
        const int row0 = m0 + (lhalf ? 8 : 0);
        for (int j = 0; j < 8; ++j) {
            for (int i = 0; i < 8; ++i) {
                const int tt = row0 + i;
                const int bb = sid[tt];
                const float* hr = H  + (size_t)tt * DD;
                const float* cr = Cp + (size_t)tt * DD;
                const float* fr = F  + (size_t)bb * DD;
                const float* bc = W + (j * 16 + lmod);
                float s = 0.f;
                for (int k = 0; k < DD; ++k)
                    s += hr[k] * bc[(size_t)k * NSLOTS];
                for (int k = 0; k < DD; ++k)
                    s += fr[k] * cr[k] * bc[(size_t)(DD + k) * NSLOTS];
                acc[j][i] = s;
            }
        }
    }
#endif
    const int row0 = m0 + (lhalf ? 8 : 0);
#pragma unroll
    for (int j = 0; j < 8; ++j) {
        float* cout = C + (size_t)row0 * NSLOTS + (j * 16 + lmod);
#pragma unroll
        for (int i = 0; i < 8; ++i) cout[(size_t)i * NSLOTS] = acc[j][i];
    }
}

// ---------------------------------------------------------------------------
// Precompute: v1 = W_V1 · w_ID,  v2 = W_V2 · w_ID,  bdot = b_V2 · w_ID,
// starts[b] = exclusive prefix sum of sent_seqs.
// ---------------------------------------------------------------------------
__global__ void k_prep(const float* __restrict__ WV1,
                       const float* __restrict__ WV2,
                       const float* __restrict__ bV2,
                       const float* __restrict__ wID,
                       const int*   __restrict__ seqs,
                       float* __restrict__ v1, float* __restrict__ v2,
                       float* __restrict__ bdot, int* __restrict__ starts) {
    const int gid = blockIdx.x * blockDim.x + threadIdx.x;
    if (gid < DD) {
        const float* r = WV1 + (size_t)gid * DD;
        float acc = 0.f;
        for (int j = 0; j < DD; ++j) acc += r[j] * wID[j];
        v1[gid] = acc;
    } else if (gid < 2 * DD) {
        const int i = gid - DD;
        const float* r = WV2 + (size_t)i * DD;
        float acc = 0.f;
        for (int j = 0; j < DD; ++j) acc += r[j] * wID[j];
        v2[i] = acc;
    } else if (gid == 2 * DD) {
        float acc = 0.f;
        for (int j = 0; j < DD; ++j) acc += bV2[j] * wID[j];
        *bdot = acc;
    } else if (gid == 2 * DD + 1) {
        int acc = 0;
        for (int b = 0; b < BB; ++b) { starts[b] = acc; acc += seqs[b]; }
    }
}

// hidID[t] = H[t,:]·v2 + bdot        (one wave per token)
__global__ void k_hid(const float* __restrict__ H, const float* __restrict__ v2,
                      const float* __restrict__ bdot, float* __restrict__ hidID) {
    const int lane = threadIdx.x & 31;
    const int t    = blockIdx.x * 8 + (threadIdx.x >> 5);
    const float* hrow = H + (size_t)t * DD;
    float acc = 0.f;
    for (int j = lane; j < DD; j += 32) acc += hrow[j] * v2[j];
    acc = wave_reduce_sum(acc);
    if (lane == 0) hidID[t] = acc + *bdot;
}

// g[b,d] = Σ_tokens tanh(Cp[t,d] + s[b,d])   (one block per sentence)
__global__ void k_g(const float* __restrict__ Cp, const float* __restrict__ s,
                    const int* __restrict__ starts, const int* __restrict__ seqs,
                    float* __restrict__ g) {
    const int b  = blockIdx.x;
    const int st = starts[b], len = seqs[b];
    for (int d = threadIdx.x; d < DD; d += blockDim.x) {
        const float sb = s[(size_t)b * DD + d];
        const float* p = Cp + (size_t)st * DD + d;
        float acc = 0.f;
        for (int i = 0; i < len; ++i) acc += tanhf(p[(size_t)i * DD] + sb);
        g[(size_t)b * DD + d] = acc;
    }
}

// e[t] = Σ_d F[b,d]*Cp[t,d]*v1[d] + hidID[t]   (one wave per token)
__global__ void k_e(const float* __restrict__ Cp, const float* __restrict__ F,
                    const float* __restrict__ v1, const float* __restrict__ hidID,
                    const int* __restrict__ sid, float* __restrict__ e) {
    const int lane = threadIdx.x & 31;
    const int t    = blockIdx.x * 8 + (threadIdx.x >> 5);
    const int b    = sid[t];
    const float* crow = Cp + (size_t)t * DD;
    const float* frow = F  + (size_t)b * DD;
    float acc = 0.f;
    for (int j = lane; j < DD; j += 32) acc += frow[j] * crow[j] * v1[j];
    acc = wave_reduce_sum(acc);
    if (lane == 0) e[t] = acc + hidID[t];
}

// softmax over each ragged sentence (one 128-thread block per sentence)
__global__ void k_softmax(const float* __restrict__ e, const int* __restrict__ starts,
                          const int* __restrict__ seqs, float* __restrict__ alpha) {
    __shared__ float sm[LL];
    const int b = blockIdx.x, tid = threadIdx.x;
    const int st = starts[b], len = seqs[b];
    const float x = (tid < len) ? e[st + tid] : -3.0e38f;
    sm[tid] = x; __syncthreads();
    for (int o = 64; o > 0; o >>= 1) {
        if (tid < o) sm[tid] = fmaxf(sm[tid], sm[tid + o]);
        __syncthreads();
    }
    const float mx = sm[0]; __syncthreads();
    const float ex = (tid < len) ? __expf(x - mx) : 0.f;
    sm[tid] = ex; __syncthreads();
    for (int o = 64; o > 0; o >>= 1) {
        if (tid < o) sm[tid] += sm[tid + o];
        __syncthreads();
    }
    const float inv = 1.f / sm[0];
    if (tid < len) alpha[st + tid] = ex * inv;
}

// r_inte[b,d] += F[b,d]*Σ α·Cp + Σ (1-α)·tanh(H)   (one block per sentence)
__global__ void k_upd(const float* __restrict__ Cp, const float* __restrict__ H,
                      const float* __restrict__ F, const float* __restrict__ alpha,
                      const int* __restrict__ starts, const int* __restrict__ seqs,
                      float* __restrict__ r_inte) {
    const int b  = blockIdx.x;
    const int st = starts[b], len = seqs[b];
    for (int d = threadIdx.x; d < DD; d += blockDim.x) {
        float a1 = 0.f, a2 = 0.f;
        const float* cp = Cp + (size_t)st * DD + d;
        const float* hp = H  + (size_t)st * DD + d;
        for (int i = 0; i < len; ++i) {
            const float al = alpha[st + i];
            a1 += al * cp[(size_t)i * DD];
            a2 += (1.f - al) * tanhf(hp[(size_t)i * DD]);
        }
        const size_t idx = (size_t)b * DD + d;
        r_inte[idx] += F[idx] * a1 + a2;
    }
}

// ---------------------------------------------------------------------------
// Host-side orchestration
// ---------------------------------------------------------------------------
extern "C" void kernel_launch(void* const* d_in, const int* in_sizes, int n_in,
                              void* d_out, int out_size, void* d_ws, size_t ws_size,
                              hipStream_t stream) {
    const float* H      = (const float*)d_in[0];
    const float* c_inte = (const float*)d_in[1];
    const float* Cp     = (const float*)d_in[2];   // c_slot_cont, flat [T,D]
    const int*   seqs   = (const int*)  d_in[3];
    const int*   sid    = (const int*)  d_in[4];
    // d_in[5] pos_in_sent unused (flat ragged layout already ordered)
    const float* W_SF   = (const float*)d_in[6];
    const float* W_VSF  = (const float*)d_in[7];
    const float* W_V1   = (const float*)d_in[8];
    const float* W_V2   = (const float*)d_in[9];
    const float* b_V2   = (const float*)d_in[10];
    const float* w_ID   = (const float*)d_in[11];
    const float* W_inte = (const float*)d_in[12];
    const float* W_slot = (const float*)d_in[13];
    float* out = (float*)d_out;

    // workspace carve-out (all offsets 16B aligned)
    char* wsb = (char*)d_ws;
    float* v1     = (float*)(wsb + 0);            //  768 f
    float* v2     = (float*)(wsb + 3072);         //  768 f
    float* bdot   = (float*)(wsb + 6144);         //    1 f
    int*   starts = (int*)  (wsb + 6400);         //  256 i
    float* hidID  = (float*)(wsb + 7680);         //    T f
    float* r_inte = (float*)(wsb + 89600);        //  B*D f
    float* s      = (float*)(wsb + 876032);       //  B*D f
    float* g      = (float*)(wsb + 1662464);      //  B*D f
    float* F      = (float*)(wsb + 2448896);      //  B*D f
    float* e      = (float*)(wsb + 3235328);      //    T f
    float* alpha  = (float*)(wsb + 3317248);      //    T f  (end 3399168 B)

    // r_inte <- c_inte
    hipMemcpyAsync(r_inte, c_inte, (size_t)BB * DD * sizeof(float),
                   hipMemcpyDeviceToDevice, stream);

    k_prep<<<7, 256, 0, stream>>>(W_V1, W_V2, b_V2, w_ID, seqs, v1, v2, bdot, starts);
    k_hid <<<TT / 8, 256, 0, stream>>>(H, v2, bdot, hidID);

    for (int it = 0; it < ITERS; ++it) {
        // s = r_inte @ W_SF   [256,768]x[768,768]: 16 x 12 wave-strips = 192
        gemm_wmma_f32<4><<<24, 256, 0, stream>>>(r_inte, W_SF, s, BB, DD, DD);
        k_g<<<BB, 256, 0, stream>>>(Cp, s, starts, seqs, g);
        // F = g @ W_VSF       [256,768]x[768,768]
        gemm_wmma_f32<4><<<24, 256, 0, stream>>>(g, W_VSF, F, BB, DD, DD);
        k_e<<<TT / 8, 256, 0, stream>>>(Cp, F, v1, hidID, sid, e);
        k_softmax<<<BB, 128, 0, stream>>>(e, starts, seqs, alpha);
        k_upd<<<BB, 256, 0, stream>>>(Cp, H, F, alpha, starts, seqs, r_inte);
    }

    // intent_output = r_inte @ W_inte  [256,768]x[768,32]: 16 x 1 strips = 16
    gemm_wmma_f32<2><<<2, 256, 0, stream>>>(r_inte, W_inte, out, BB, NLABELS, DD);
    // slot_output = [H | F⊙Cp] @ W_slot: 1280 x 1 (16x128) strips = 1280 waves
    gemm_slot_wmma<<<160, 256, 0, stream>>>(H, Cp, F, sid, W_slot, out + BB * NLABELS);
}